// CenterPointPostProcess_83021717832200
// MI455X (gfx1250) — compile-verified
//
#include <hip/hip_runtime.h>
#include <stdint.h>

// ---------------- problem constants ----------------
#define BB 4
#define CC 10
#define HH 512
#define WW 512
#define HWN (HH*WW)          // 262144
#define NN  (CC*HWN)         // 2621440 elements per batch in heatmap
#define KK  1000
#define KP  1024             // padded K
#define CAP 4096             // candidate capacity per batch (expected ~1-2K)
#define BINS 4096            // 12-bit histogram
#define POSTMAX 83
#define R2 16.0f             // RADIUS^2

typedef float v2f __attribute__((ext_vector_type(2)));
typedef float v8f __attribute__((ext_vector_type(8)));

// order-preserving float -> uint key (larger float => larger key)
__device__ __forceinline__ unsigned flipf(float f) {
  unsigned u = __float_as_uint(f);
  return (u & 0x80000000u) ? ~u : (u | 0x80000000u);
}
__device__ __forceinline__ float unflipf(unsigned u) {
  unsigned v = (u & 0x80000000u) ? (u & 0x7FFFFFFFu) : ~u;
  return __uint_as_float(v);
}

// ---------------- workspace layout (in 4-byte words) ----------------
enum : size_t {
  OFF_HIST  = 0,                         // BB*BINS u32
  OFF_T     = OFF_HIST + BB*BINS,        // BB u32
  OFF_CNT   = OFF_T + BB,                // BB u32
  OFF_CANDV = OFF_CNT + BB,              // BB*CAP f32
  OFF_CANDI = OFF_CANDV + BB*CAP,        // BB*CAP u32
  OFF_TVAL  = OFF_CANDI + BB*CAP,        // BB*KP f32 (logits)
  OFF_TIDX  = OFF_TVAL + BB*KP,          // BB*KP u32
  OFF_NRM   = OFF_TIDX + BB*KP,          // BB*KP f32 (x^2+y^2)
  OFF_CPK   = OFF_NRM + BB*KP,           // BB*KP v2f (packed centers, 8B aligned)
  OFF_SC    = OFF_CPK + BB*KP*2,         // BB*KP f32 (sigmoid scores)
  OFF_LB    = OFF_SC + BB*KP,            // BB*KP f32 (labels)
  OFF_BBOX  = OFF_LB + BB*KP,            // BB*KP*9 f32 (unmasked boxes)
  OFF_SUP   = OFF_BBOX + BB*KP*9,        // BB*KP*32 u32 suppression bitmask
  OFF_KEEP  = OFF_SUP + BB*KP*32,        // BB*32 u32 keep bitmask
  OFF_END   = OFF_KEEP + BB*32
};

// ---------------- kernel 0: zero hist + counters ----------------
__global__ void __launch_bounds__(256) zero_kernel(uint32_t* hist, uint32_t* cnt) {
  int g = blockIdx.x * 256 + threadIdx.x;
  if (g < BB*BINS) hist[g] = 0u;
  if (g < BB) cnt[g] = 0u;
}

// ---------------- kernel 1: 12-bit histogram of logits per batch ----------------
__global__ void __launch_bounds__(256) hist_kernel(const float* __restrict__ hm,
                                                   uint32_t* __restrict__ hist) {
  __shared__ uint32_t lh[BINS];
  const int b = blockIdx.y;
  for (int i = threadIdx.x; i < BINS; i += 256) lh[i] = 0u;
  __syncthreads();
  const size_t base = (size_t)b * NN + (size_t)blockIdx.x * 4096;
  #pragma unroll 4
  for (int j = 0; j < 16; ++j) {
    float f = hm[base + threadIdx.x + j * 256];
    atomicAdd(&lh[flipf(f) >> 20], 1u);
  }
  __syncthreads();
  for (int i = threadIdx.x; i < BINS; i += 256) {
    uint32_t v = lh[i];
    if (v) atomicAdd(&hist[b*BINS + i], v);
  }
}

// ---------------- kernel 2: find threshold bin per batch ----------------
__global__ void thresh_kernel(const uint32_t* __restrict__ hist, uint32_t* __restrict__ T) {
  int b = blockIdx.x;
  if (threadIdx.x == 0) {
    unsigned acc = 0; int t = 0;
    for (int i = BINS - 1; i >= 0; --i) {
      acc += hist[b*BINS + i];
      if (acc >= KK) { t = i; break; }
    }
    T[b] = (unsigned)t;
  }
}

// ---------------- kernel 3: collect candidates >= threshold bin ----------------
__global__ void __launch_bounds__(256) collect_kernel(const float* __restrict__ hm,
                                                      const uint32_t* __restrict__ T,
                                                      uint32_t* __restrict__ cnt,
                                                      float* __restrict__ candv,
                                                      uint32_t* __restrict__ candi) {
  const int b = blockIdx.y;
  const unsigned t = T[b];
  const size_t base = (size_t)blockIdx.x * 4096;
  #pragma unroll 4
  for (int j = 0; j < 16; ++j) {
    size_t idx = base + threadIdx.x + j * 256;
    float f = hm[(size_t)b * NN + idx];
    if ((flipf(f) >> 20) >= t) {
      unsigned pos = atomicAdd(&cnt[b], 1u);
      if (pos < CAP) { candv[b*CAP + pos] = f; candi[b*CAP + pos] = (uint32_t)idx; }
    }
  }
}

// ---------------- kernel 4: exact top-K via bitonic sort in LDS ----------------
// composite key = (~flipf(val) << 32) | idx ; ascending sort => (val desc, idx asc)
__global__ void __launch_bounds__(256) sort_kernel(const float* __restrict__ candv,
                                                   const uint32_t* __restrict__ candi,
                                                   const uint32_t* __restrict__ cnt,
                                                   float* __restrict__ tval,
                                                   uint32_t* __restrict__ tidx) {
  __shared__ unsigned long long keys[CAP];
  const int b = blockIdx.x;
  unsigned M = cnt[b]; if (M > CAP) M = CAP;
  for (int i = threadIdx.x; i < CAP; i += 256) {
    unsigned long long key = ~0ull;
    if (i < (int)M) {
      unsigned u = flipf(candv[b*CAP + i]);
      key = ((unsigned long long)(~u) << 32) | (unsigned long long)candi[b*CAP + i];
    }
    keys[i] = key;
  }
  __syncthreads();
  for (unsigned k = 2; k <= CAP; k <<= 1) {
    for (unsigned j = k >> 1; j > 0; j >>= 1) {
      for (unsigned i = threadIdx.x; i < CAP; i += 256) {
        unsigned ixj = i ^ j;
        if (ixj > i) {
          unsigned long long a = keys[i], c2 = keys[ixj];
          bool up = ((i & k) == 0);
          if ((a > c2) == up) { keys[i] = c2; keys[ixj] = a; }
        }
      }
      __syncthreads();
    }
  }
  for (int k2 = threadIdx.x; k2 < KK; k2 += 256) {
    unsigned long long key = keys[k2];
    tval[b*KP + k2] = unflipf(~(unsigned)(key >> 32));
    tidx[b*KP + k2] = (unsigned)(key & 0xFFFFFFFFu);
  }
}

// ---------------- kernel 5: gather + decode boxes, centers, norms ----------------
__global__ void __launch_bounds__(256) decode_kernel(const float* __restrict__ reg,
                                                     const float* __restrict__ height,
                                                     const float* __restrict__ dimq,
                                                     const float* __restrict__ rot,
                                                     const float* __restrict__ vel,
                                                     const float* __restrict__ tval,
                                                     const uint32_t* __restrict__ tidx,
                                                     v2f* __restrict__ cpk,
                                                     float* __restrict__ nrm,
                                                     float* __restrict__ sc, float* __restrict__ lb,
                                                     float* __restrict__ bbox) {
  int g = blockIdx.x * 256 + threadIdx.x;   // 0 .. BB*KP-1
  if (g >= BB*KP) return;
  int b = g / KP, k = g % KP;
  if (k >= KK) { v2f z = {0.f, 0.f}; cpk[g] = z; nrm[g] = 0.f; return; }
  float logit = tval[g];
  uint32_t idx = tidx[g];
  int spatial = (int)(idx % HWN);
  int label   = (int)(idx / HWN);
  float fx = (float)(spatial % WW);
  float fy = (float)(spatial / WW);
  size_t sb2 = (size_t)b * 2 * HWN;
  float r0 = reg[sb2 + spatial];
  float r1 = reg[sb2 + HWN + spatial];
  float xs = (fx + r0) * 0.8f - 51.2f;      // OUT_SIZE_FACTOR*VOXEL=0.8, PC_MIN=-51.2
  float ys = (fy + r1) * 0.8f - 51.2f;
  float hei = height[(size_t)b * HWN + spatial];
  size_t sb3 = (size_t)b * 3 * HWN;
  float d0 = expf(dimq[sb3 + spatial]);
  float d1 = expf(dimq[sb3 + HWN + spatial]);
  float d2 = expf(dimq[sb3 + 2*HWN + spatial]);
  float rs = rot[sb2 + spatial];
  float rc = rot[sb2 + HWN + spatial];
  float ang = atan2f(rs, rc);
  float v0 = vel[sb2 + spatial];
  float v1 = vel[sb2 + HWN + spatial];
  v2f c; c.x = xs; c.y = ys;
  cpk[g] = c;
  nrm[g] = xs*xs + ys*ys;
  sc[g] = 1.0f / (1.0f + expf(-logit));
  lb[g] = (float)label;
  float* bp = bbox + (size_t)g * 9;
  bp[0]=xs; bp[1]=ys; bp[2]=hei; bp[3]=d0; bp[4]=d1; bp[5]=d2; bp[6]=ang; bp[7]=v0; bp[8]=v1;
}

// ---------------- kernel 6: pairwise-distance suppression matrix via WMMA ----------------
// d2(i,j) = n_i + n_j - 2*(x_i*x_j + y_i*y_j); cross term via V_WMMA_F32_16X16X4_F32
// A: 16x4 (rows = centers i, K0=x,K1=y,K2=K3=0), B: 4x16 (cols = centers j).
// Loads are unconditional (all indices valid); lane>=16 operands are zeroed by select.
// No row/col bounds checks needed: bits for cols>=1000 only clear keep bits that start
// at 0, and rows>=1000 are never read by the NMS loop.
__global__ void __launch_bounds__(32) supmat_kernel(const v2f* __restrict__ cpk,
                                                    const float* __restrict__ nrm,
                                                    uint16_t* __restrict__ sup16) {
  const int ti = blockIdx.x;     // 0..63  (i-tile)
  const int b  = blockIdx.y;
  const int lane = threadIdx.x;  // 0..31, wave32
  const bool lo = (lane < 16);
  const v2f*   cb = cpk + b * KP;
  const float* nb = nrm + b * KP;

  // A operand: lanes 0-15 hold K=0 (x) in VGPR0 and K=1 (y) in VGPR1; lanes 16-31 hold K=2,3 = 0
  v2f av = cb[ti*16 + (lane & 15)];
  v2f a;
  a.x = lo ? av.x : 0.0f;
  a.y = lo ? av.y : 0.0f;

  // C/D row layout: VGPR v -> row = ti*16 + v + 8*(lane>=16)
  const int rbase = ti*16 + ((lane >> 4) << 3);
  float ni[8];
  #pragma unroll
  for (int v = 0; v < 8; ++v) ni[v] = nb[rbase + v];

  for (int tj = 0; tj < 64; ++tj) {
    const int col = tj*16 + (lane & 15);
    v2f cv = cb[col];                       // unconditional b64 load
    v2f bv;
    bv.x = lo ? cv.x : 0.0f;                // B row K=0
    bv.y = lo ? cv.y : 0.0f;                // B row K=1
    const float nj = nb[col];
    v8f acc = {};
    v8f d = __builtin_amdgcn_wmma_f32_16x16x4_f32(
        /*neg_a=*/false, a, /*neg_b=*/false, bv,
        /*c_mod=*/(short)0, acc, /*reuse_a=*/false, /*reuse_b=*/false);
    unsigned ms[8];
    #pragma unroll
    for (int v = 0; v < 8; ++v) {
      ms[v] = __builtin_amdgcn_ballot_w32(ni[v] + nj - 2.0f * d[v] <= R2);
    }
    if (lane == 0) {
      #pragma unroll
      for (int v = 0; v < 8; ++v) {
        // ballot bits [15:0] = row rbase+v (lanes 0-15), bits [31:16] = row rbase+v+8
        sup16[((size_t)(b*KP + ti*16 + v))     * 64 + tj] = (uint16_t)(ms[v] & 0xFFFFu);
        sup16[((size_t)(b*KP + ti*16 + v + 8)) * 64 + tj] = (uint16_t)(ms[v] >> 16);
      }
    }
  }
}

// ---------------- kernel 7: sequential circle-NMS on bitmasks + POST_MAX ----------------
// One block per batch. Stage 128KB suppression matrix in LDS (WGP has 320KB),
// then wave 0 runs the serial loop: lane l owns keep bits [32l, 32l+32).
__global__ void __launch_bounds__(256) nms_kernel(const uint32_t* __restrict__ sup,
                                                  uint32_t* __restrict__ keepw) {
  extern __shared__ uint32_t s[];
  const int b = blockIdx.x;
  const uint32_t* srow = sup + (size_t)b * KP * 32;
  for (int i = threadIdx.x; i < KP*32; i += 256) s[i] = srow[i];
  __syncthreads();
  if (threadIdx.x < 32) {
    const int lane = threadIdx.x;
    uint32_t keep = (lane < 31) ? 0xFFFFFFFFu : 0x000000FFu;  // 1000 = 31*32 + 8
    for (int i = 0; i < KK; ++i) {
      const int w = i >> 5, t = i & 31;
      const uint32_t kw = (uint32_t)__builtin_amdgcn_ds_bpermute(w << 2, (int)keep);
      if ((kw >> t) & 1u) {                 // uniform branch (broadcast value)
        uint32_t row = s[i*32 + lane];
        uint32_t hm = (t == 31) ? 0u : (0xFFFFFFFFu << (t + 1));  // only j > i
        uint32_t msk = (lane > w) ? 0xFFFFFFFFu : (lane == w ? hm : 0u);
        keep &= ~(row & msk);
      }
    }
    // POST_MAX: keep only first 83 kept (cumsum <= 83)
    unsigned cnt = __popc(keep);
    unsigned inc = cnt;
    #pragma unroll
    for (int off = 1; off < 32; off <<= 1) {
      unsigned n = (unsigned)__builtin_amdgcn_ds_bpermute(((lane - off) & 31) << 2, (int)inc);
      if (lane >= off) inc += n;
    }
    unsigned c = inc - cnt;   // exclusive prefix of kept count
    uint32_t nk = 0u;
    for (int t2 = 0; t2 < 32; ++t2) {
      if ((keep >> t2) & 1u) { if (c < POSTMAX) nk |= (1u << t2); c++; }
    }
    keepw[b*32 + lane] = nk;
  }
}

// ---------------- kernel 8: masked final write ----------------
// Output layout (floats): bboxes[B,K,9] | scores[B,K] | labels[B,K] | keep[B,K]
__global__ void __launch_bounds__(256) final_kernel(const float* __restrict__ bbox,
                                                    const float* __restrict__ sc,
                                                    const float* __restrict__ lb,
                                                    const uint32_t* __restrict__ keepw,
                                                    float* __restrict__ out) {
  int g = blockIdx.x * 256 + threadIdx.x;
  if (g >= BB*KK) return;
  int b = g / KK, k = g % KK;
  uint32_t w = keepw[b*32 + (k >> 5)];
  float m = (float)((w >> (k & 31)) & 1u);
  const float* bp = bbox + ((size_t)(b*KP + k)) * 9;
  float* ob = out + (size_t)g * 9;
  #pragma unroll
  for (int j = 0; j < 9; ++j) ob[j] = bp[j] * m;
  out[BB*KK*9  + g] = sc[b*KP + k] * m;
  out[BB*KK*10 + g] = lb[b*KP + k];     // labels are NOT masked in the reference
  out[BB*KK*11 + g] = m;                // keep flag
}

// ---------------- launch ----------------
extern "C" void kernel_launch(void* const* d_in, const int* in_sizes, int n_in,
                              void* d_out, int out_size, void* d_ws, size_t ws_size,
                              hipStream_t stream) {
  const float* hm     = (const float*)d_in[0];
  const float* reg    = (const float*)d_in[1];
  const float* height = (const float*)d_in[2];
  const float* dimq   = (const float*)d_in[3];
  const float* rot    = (const float*)d_in[4];
  const float* vel    = (const float*)d_in[5];
  float* out = (float*)d_out;

  uint32_t* W = (uint32_t*)d_ws;
  uint32_t* hist  = W + OFF_HIST;
  uint32_t* T     = W + OFF_T;
  uint32_t* cnt   = W + OFF_CNT;
  float*    candv = (float*)(W + OFF_CANDV);
  uint32_t* candi = W + OFF_CANDI;
  float*    tval  = (float*)(W + OFF_TVAL);
  uint32_t* tidx  = W + OFF_TIDX;
  float*    nrm   = (float*)(W + OFF_NRM);
  v2f*      cpk   = (v2f*)(W + OFF_CPK);
  float*    sc    = (float*)(W + OFF_SC);
  float*    lb    = (float*)(W + OFF_LB);
  float*    bbox  = (float*)(W + OFF_BBOX);
  uint32_t* sup   = W + OFF_SUP;
  uint16_t* sup16 = (uint16_t*)(W + OFF_SUP);
  uint32_t* keepw = W + OFF_KEEP;

  // 0) zero hist + counters
  zero_kernel<<<(BB*BINS + 255)/256, 256, 0, stream>>>(hist, cnt);
  // 1) histogram (pass 1 over heatmap)
  hist_kernel<<<dim3(NN/4096, BB), 256, 0, stream>>>(hm, hist);
  // 2) threshold bin per batch
  thresh_kernel<<<BB, 32, 0, stream>>>(hist, T);
  // 3) collect candidates (pass 2 over heatmap)
  collect_kernel<<<dim3(NN/4096, BB), 256, 0, stream>>>(hm, T, cnt, candv, candi);
  // 4) exact top-K ordering via bitonic sort
  sort_kernel<<<BB, 256, 0, stream>>>(candv, candi, cnt, tval, tidx);
  // 5) gather + decode
  decode_kernel<<<(BB*KP)/256, 256, 0, stream>>>(reg, height, dimq, rot, vel,
                                                 tval, tidx, cpk, nrm, sc, lb, bbox);
  // 6) suppression bitmask matrix via WMMA f32 16x16x4
  supmat_kernel<<<dim3(64, BB), 32, 0, stream>>>(cpk, nrm, sup16);
  // 7) serial circle-NMS + POST_MAX (128KB dynamic LDS)
  nms_kernel<<<BB, 256, (size_t)KP*32*4, stream>>>(sup, keepw);
  // 8) masked output write
  final_kernel<<<(BB*KK + 255)/256, 256, 0, stream>>>(bbox, sc, lb, keepw, out);
}